// SharedField_64828236366555
// MI455X (gfx1250) — compile-verified
//
#include <hip/hip_runtime.h>

typedef __attribute__((ext_vector_type(2))) float v2f;
typedef __attribute__((ext_vector_type(4))) float v4f;
typedef __attribute__((ext_vector_type(8))) float v8f;

#define BATCH 8
#define NNODE 2048
#define DDIM  1024
#define TM 128
#define TN 128
#define KC 16
#define KCA (KC + 4)     /* A tile LDS row stride: 20 floats (16B aligned, bank-clean) */
#define TNB (TN + 20)    /* B tile LDS row stride: 148 floats (16B aligned, bank-clean) */
#define EPSV 1e-8f
#define PAIR_SCALE 0.04419417382415922f   /* 1/sqrt(512) */

__device__ __forceinline__ float sigmoidf_(float x) {
    return 1.0f / (1.0f + __expf(-x));
}

// One LDS K-stage of WMMA f32 16x16x4 work: 4 k-steps x (2 M-frags x 4 N-frags).
__device__ __forceinline__ void mma_stage(const float (&As)[TM][KCA],
                                          const float (&Bs)[KC][TNB],
                                          v8f (&acc)[2][4],
                                          int wm, int wn, int lr, int hh) {
#pragma unroll
    for (int ks = 0; ks < KC; ks += 4) {
        v2f afr[2];
        v2f bfr[4];
#pragma unroll
        for (int mf = 0; mf < 2; ++mf) {
            const int rr = wm * 32 + mf * 16 + lr;          // M = lane%16 (+frag)
            afr[mf][0] = As[rr][ks + 2 * hh + 0];           // K = v + 2*(lane/16)
            afr[mf][1] = As[rr][ks + 2 * hh + 1];
        }
#pragma unroll
        for (int nf = 0; nf < 4; ++nf) {
            const int cc = wn * 64 + nf * 16 + lr;          // N = lane%16 (+frag)
            bfr[nf][0] = Bs[ks + 2 * hh + 0][cc];           // K = v + 2*(lane/16)
            bfr[nf][1] = Bs[ks + 2 * hh + 1][cc];
        }
#pragma unroll
        for (int mf = 0; mf < 2; ++mf)
#pragma unroll
            for (int nf = 0; nf < 4; ++nf)
                acc[mf][nf] = __builtin_amdgcn_wmma_f32_16x16x4_f32(
                    false, afr[mf], false, bfr[nf], (short)0, acc[mf][nf],
                    false, false);
    }
}

__device__ __forceinline__ void acc_zero(v8f (&acc)[2][4]) {
#pragma unroll
    for (int mf = 0; mf < 2; ++mf)
#pragma unroll
        for (int nf = 0; nf < 4; ++nf)
#pragma unroll
            for (int r = 0; r < 8; ++r) acc[mf][nf][r] = 0.0f;
}

// OUT(M=B*N rows, D cols) = X @ W^T, optionally gating X rows, optionally
// pair-normalizing the output (cos/sin of atan2 identity, folds 1/sqrt(pairs)).
template <bool NORM, bool GATED>
__global__ __launch_bounds__(256) void gemm_xw_kernel(
    const float* __restrict__ X, const float* __restrict__ W,
    const float* __restrict__ gate, float* __restrict__ OUT) {
    __shared__ float As[2][TM][KCA];
    __shared__ float Bs[2][KC][TNB];

    const int tid = threadIdx.x;
    const int lane = tid & 31, lr = lane & 15, hh = lane >> 4;
    const int wid = tid >> 5, wm = wid & 3, wn = wid >> 2;
    const int n0 = blockIdx.x * TN;
    const int m0 = blockIdx.y * TM;

    // loader geometry: 512 float4 per tile, 2 per thread (rows mm0 and mm0+64)
    const int mm0 = tid >> 2;
    const int k40 = (tid & 3) << 2;
    const float* ArowP0 = X + (size_t)(m0 + mm0) * DDIM + k40;
    const float* ArowP1 = X + (size_t)(m0 + mm0 + 64) * DDIM + k40;
    const float* BrowP0 = W + (size_t)(n0 + mm0) * DDIM + k40;
    const float* BrowP1 = W + (size_t)(n0 + mm0 + 64) * DDIM + k40;
    const float ga0 = GATED ? gate[m0 + mm0] : 1.0f;
    const float ga1 = GATED ? gate[m0 + mm0 + 64] : 1.0f;

    v8f acc[2][4];
    acc_zero(acc);
    v4f ra0, ra1, rb0, rb1;

#define XW_LOADG(K0)                                             \
    ra0 = *reinterpret_cast<const v4f*>(ArowP0 + (K0));          \
    ra1 = *reinterpret_cast<const v4f*>(ArowP1 + (K0));          \
    rb0 = *reinterpret_cast<const v4f*>(BrowP0 + (K0));          \
    rb1 = *reinterpret_cast<const v4f*>(BrowP1 + (K0));

#define XW_STORE(NB)                                             \
    if (GATED) { ra0 *= ga0; ra1 *= ga1; }                       \
    *reinterpret_cast<v4f*>(&As[(NB)][mm0][k40]) = ra0;          \
    *reinterpret_cast<v4f*>(&As[(NB)][mm0 + 64][k40]) = ra1;     \
    _Pragma("unroll")                                            \
    for (int c = 0; c < 4; ++c) {                                \
        Bs[(NB)][k40 + c][mm0] = rb0[c];                         \
        Bs[(NB)][k40 + c][mm0 + 64] = rb1[c];                    \
    }

    constexpr int NST = DDIM / KC;
    XW_LOADG(0);
    XW_STORE(0);
    __syncthreads();
    for (int s = 0; s < NST; ++s) {
        if (s + 1 < NST) { XW_LOADG((s + 1) * KC); }
        mma_stage(As[s & 1], Bs[s & 1], acc, wm, wn, lr, hh);
        if (s + 1 < NST) { XW_STORE((s + 1) & 1); }
        __syncthreads();
    }
#undef XW_LOADG
#undef XW_STORE

#pragma unroll
    for (int mf = 0; mf < 2; ++mf)
#pragma unroll
        for (int nf = 0; nf < 4; ++nf)
#pragma unroll
            for (int r = 0; r < 8; ++r) {
                float v = acc[mf][nf][r];
                if (NORM) {
                    // pair partner column sits in the xor-1 lane (same VGPR row)
                    float p = __shfl_xor(v, 1, 32);
                    float own = v + EPSV, oth = p + EPSV;
                    v = own * rsqrtf(own * own + oth * oth) * PAIR_SCALE;
                }
                const int row = m0 + wm * 32 + mf * 16 + r + 8 * hh;
                const int col = n0 + wn * 64 + nf * 16 + lr;
                OUT[(size_t)row * DDIM + col] = v;
            }
}

// Per batch: r_inst = Chat @ Chat^T (scale folded), fuse EMA + clip -> Rnew.
__global__ __launch_bounds__(256) void resonance_kernel(
    const float* __restrict__ Chat, const float* __restrict__ Racc,
    float* __restrict__ Rnew) {
    __shared__ float As[2][TM][KCA];
    __shared__ float Bs[2][KC][TNB];

    const int tid = threadIdx.x;
    const int lane = tid & 31, lr = lane & 15, hh = lane >> 4;
    const int wid = tid >> 5, wm = wid & 3, wn = wid >> 2;
    const int b = blockIdx.z;
    const int n0 = blockIdx.x * TN;   // column tile (sender node m)
    const int m0 = blockIdx.y * TM;   // row tile (receiver node n)
    const float* A = Chat + (size_t)b * NNODE * DDIM;

    const int mm0 = tid >> 2;
    const int k40 = (tid & 3) << 2;
    const float* ArowP0 = A + (size_t)(m0 + mm0) * DDIM + k40;
    const float* ArowP1 = A + (size_t)(m0 + mm0 + 64) * DDIM + k40;
    const float* BrowP0 = A + (size_t)(n0 + mm0) * DDIM + k40;
    const float* BrowP1 = A + (size_t)(n0 + mm0 + 64) * DDIM + k40;

    v8f acc[2][4];
    acc_zero(acc);
    v4f ra0, ra1, rb0, rb1;

#define RS_LOADG(K0)                                             \
    ra0 = *reinterpret_cast<const v4f*>(ArowP0 + (K0));          \
    ra1 = *reinterpret_cast<const v4f*>(ArowP1 + (K0));          \
    rb0 = *reinterpret_cast<const v4f*>(BrowP0 + (K0));          \
    rb1 = *reinterpret_cast<const v4f*>(BrowP1 + (K0));

#define RS_STORE(NB)                                             \
    *reinterpret_cast<v4f*>(&As[(NB)][mm0][k40]) = ra0;          \
    *reinterpret_cast<v4f*>(&As[(NB)][mm0 + 64][k40]) = ra1;     \
    _Pragma("unroll")                                            \
    for (int c = 0; c < 4; ++c) {                                \
        Bs[(NB)][k40 + c][mm0] = rb0[c];                         \
        Bs[(NB)][k40 + c][mm0 + 64] = rb1[c];                    \
    }

    constexpr int NST = DDIM / KC;
    RS_LOADG(0);
    RS_STORE(0);
    __syncthreads();
    for (int s = 0; s < NST; ++s) {
        if (s + 1 < NST) { RS_LOADG((s + 1) * KC); }
        mma_stage(As[s & 1], Bs[s & 1], acc, wm, wn, lr, hh);
        if (s + 1 < NST) { RS_STORE((s + 1) & 1); }
        __syncthreads();
    }
#undef RS_LOADG
#undef RS_STORE

#pragma unroll
    for (int mf = 0; mf < 2; ++mf)
#pragma unroll
        for (int nf = 0; nf < 4; ++nf)
#pragma unroll
            for (int r = 0; r < 8; ++r) {
                const int row = m0 + wm * 32 + mf * 16 + r + 8 * hh;
                const int col = n0 + wn * 64 + nf * 16 + lr;
                const size_t idx = (size_t)(b * NNODE + row) * NNODE + col;
                float v = 0.7f * Racc[idx] + 0.3f * acc[mf][nf][r];
                v = fminf(fmaxf(v, -2.0f), 2.0f);
                Rnew[idx] = v;
            }
}

// scale[b,n] = gate_n / (gate_n * sum_m sigmoid(c*r[n,m])*(m!=n)*gate_m + 1e-8)
__global__ __launch_bounds__(256) void rowscale_kernel(
    const float* __restrict__ Rnew, const float* __restrict__ gate,
    const float* __restrict__ condp, float* __restrict__ scale) {
    const int n = blockIdx.x, b = blockIdx.y;
    const float cond = fminf(fmaxf(condp[0], -5.0f), 5.0f);
    const float* row = Rnew + (size_t)(b * NNODE + n) * NNODE;
    const float* g = gate + b * NNODE;

    float sum = 0.0f;
    for (int m4 = threadIdx.x * 4; m4 < NNODE; m4 += 1024) {
        const v4f r = *reinterpret_cast<const v4f*>(&row[m4]);
        const v4f gg = *reinterpret_cast<const v4f*>(&g[m4]);
#pragma unroll
        for (int c = 0; c < 4; ++c) {
            float s = sigmoidf_(cond * r[c]);
            if (m4 + c == n) s = 0.0f;
            sum += s * gg[c];
        }
    }
    for (int off = 16; off > 0; off >>= 1) sum += __shfl_xor(sum, off, 32);
    __shared__ float wsum[8];
    if ((threadIdx.x & 31) == 0) wsum[threadIdx.x >> 5] = sum;
    __syncthreads();
    if (threadIdx.x == 0) {
        float t = 0.0f;
        for (int i = 0; i < 8; ++i) t += wsum[i];
        const float gn = g[n];
        scale[b * NNODE + n] = gn / (gn * t + 1e-8f);
    }
}

// field = rownorm(coupling) @ values; coupling regenerated from Rnew on load.
__global__ __launch_bounds__(256) void field_kernel(
    const float* __restrict__ Rnew, const float* __restrict__ values,
    const float* __restrict__ gate, const float* __restrict__ condp,
    const float* __restrict__ scale, float* __restrict__ field) {
    __shared__ float As[2][TM][KCA];
    __shared__ float Bs[2][KC][TNB];

    const int tid = threadIdx.x;
    const int lane = tid & 31, lr = lane & 15, hh = lane >> 4;
    const int wid = tid >> 5, wm = wid & 3, wn = wid >> 2;
    const int b = blockIdx.z;
    const int c0 = blockIdx.x * TN;   // output feature tile
    const int n0 = blockIdx.y * TM;   // receiver node tile
    const float cond = fminf(fmaxf(condp[0], -5.0f), 5.0f);
    const float* R = Rnew + (size_t)b * NNODE * NNODE;
    const float* V = values + (size_t)b * NNODE * DDIM;
    const float* g = gate + b * NNODE;

    // A loader: rows ann0/ann0+64 of R, float4 along m; transform on the fly
    const int ann0 = tid >> 2;
    const int ak40 = (tid & 3) << 2;
    const float* ArowP0 = R + (size_t)(n0 + ann0) * NNODE + ak40;
    const float* ArowP1 = R + (size_t)(n0 + ann0 + 64) * NNODE + ak40;
    // B loader: values tile, float4 along d; rows bkk0 and bkk0+8
    const int bkk0 = tid >> 5;
    const int bdd0 = (tid & 31) << 2;
    const float* BrowP0 = V + (size_t)bkk0 * DDIM + c0 + bdd0;
    const float* BrowP1 = V + (size_t)(bkk0 + 8) * DDIM + c0 + bdd0;

    v8f acc[2][4];
    acc_zero(acc);
    v4f ta0, ta1, rbv0, rbv1;

#define FD_LOADG(K0)                                                        \
    {                                                                       \
        const v4f r0 = *reinterpret_cast<const v4f*>(ArowP0 + (K0));        \
        const v4f r1 = *reinterpret_cast<const v4f*>(ArowP1 + (K0));        \
        const v4f gm = *reinterpret_cast<const v4f*>(&g[(K0) + ak40]);      \
        _Pragma("unroll")                                                   \
        for (int c = 0; c < 4; ++c) {                                       \
            const int m = (K0) + ak40 + c;                                  \
            float s0 = sigmoidf_(cond * r0[c]);                             \
            float s1 = sigmoidf_(cond * r1[c]);                             \
            if (m == n0 + ann0) s0 = 0.0f;                                  \
            if (m == n0 + ann0 + 64) s1 = 0.0f;                             \
            ta0[c] = s0 * gm[c];                                            \
            ta1[c] = s1 * gm[c];                                            \
        }                                                                   \
        rbv0 = *reinterpret_cast<const v4f*>(BrowP0 + (size_t)(K0) * DDIM); \
        rbv1 = *reinterpret_cast<const v4f*>(BrowP1 + (size_t)(K0) * DDIM); \
    }

#define FD_STORE(NB)                                                        \
    *reinterpret_cast<v4f*>(&As[(NB)][ann0][ak40]) = ta0;                   \
    *reinterpret_cast<v4f*>(&As[(NB)][ann0 + 64][ak40]) = ta1;              \
    *reinterpret_cast<v4f*>(&Bs[(NB)][bkk0][bdd0]) = rbv0;                  \
    *reinterpret_cast<v4f*>(&Bs[(NB)][bkk0 + 8][bdd0]) = rbv1;

    constexpr int NST = NNODE / KC;
    FD_LOADG(0);
    FD_STORE(0);
    __syncthreads();
    for (int s = 0; s < NST; ++s) {
        if (s + 1 < NST) { FD_LOADG((s + 1) * KC); }
        mma_stage(As[s & 1], Bs[s & 1], acc, wm, wn, lr, hh);
        if (s + 1 < NST) { FD_STORE((s + 1) & 1); }
        __syncthreads();
    }
#undef FD_LOADG
#undef FD_STORE

#pragma unroll
    for (int mf = 0; mf < 2; ++mf)
#pragma unroll
        for (int nf = 0; nf < 4; ++nf)
#pragma unroll
            for (int r = 0; r < 8; ++r) {
                const int row = n0 + wm * 32 + mf * 16 + r + 8 * hh;
                const int col = c0 + wn * 64 + nf * 16 + lr;
                field[(size_t)(b * NNODE + row) * DDIM + col] =
                    acc[mf][nf][r] * scale[b * NNODE + row];
            }
}

extern "C" void kernel_launch(void* const* d_in, const int* in_sizes, int n_in,
                              void* d_out, int out_size, void* d_ws, size_t ws_size,
                              hipStream_t stream) {
    (void)in_sizes; (void)n_in; (void)out_size; (void)ws_size;
    const float* X     = (const float*)d_in[0];   // node_states (B,N,D)
    const float* gates = (const float*)d_in[1];   // node_gates (B,N,1)
    const float* Racc  = (const float*)d_in[2];   // r_acc (B,N,N)
    const float* Win   = (const float*)d_in[3];   // W_in (D,D)
    const float* Wout  = (const float*)d_in[4];   // W_out (D,D)
    const float* condp = (const float*)d_in[5];   // conductance scalar

    float* out   = (float*)d_out;
    float* field = out;                                // (B,N,D)
    float* rnew  = out + (size_t)BATCH * NNODE * DDIM; // (B,N,N)

    float* ws     = (float*)d_ws;
    float* Chat   = ws;                                    // (B,N,D) normalized phases
    float* values = Chat + (size_t)BATCH * NNODE * DDIM;   // (B,N,D)
    float* scale  = values + (size_t)BATCH * NNODE * DDIM; // (B,N)

    const dim3 blk(256);
    gemm_xw_kernel<true, false><<<dim3(DDIM / TN, (BATCH * NNODE) / TM), blk, 0,
                                  stream>>>(X, Win, nullptr, Chat);
    gemm_xw_kernel<false, true><<<dim3(DDIM / TN, (BATCH * NNODE) / TM), blk, 0,
                                  stream>>>(X, Wout, gates, values);
    resonance_kernel<<<dim3(NNODE / TN, NNODE / TM, BATCH), blk, 0, stream>>>(
        Chat, Racc, rnew);
    rowscale_kernel<<<dim3(NNODE, BATCH), blk, 0, stream>>>(rnew, gates, condp,
                                                            scale);
    field_kernel<<<dim3(DDIM / TN, NNODE / TM, BATCH), blk, 0, stream>>>(
        rnew, values, gates, condp, scale, field);
}